// ModuleSepConv_21268678050489
// MI455X (gfx1250) — compile-verified
//
#include <hip/hip_runtime.h>

// ---------------------------------------------------------------------------
// SepConv for MI455X (gfx1250):
//   Out[n,c,y,x] = sum_{fy,fx} In[n,c,y+fy,x+fx] * Ver[n,fy,y,x] * Hor[n,fx,y,x]
//
// For fixed (n,y), Hor is independent of (c,fy), so the horizontal contraction
// for all 153 (c,fy) columns is one banded matmul
//     T[x,(c,fy)] = sum_{x'} HorBand[x,x'] * In[c, y+fy, x']
// computed with V_WMMA_F32_16X16X4_F32 (fp32 precision preserved), followed by
// a cheap per-pixel vertical reduction with Ver.
//
// v3: double-buffered sIn with software pipelining -- global loads for N-tile
// nt+1 issue before the WMMA chain of nt (latency hidden by the matrix pipe),
// LDS stores after it, one barrier per iteration.
// ---------------------------------------------------------------------------

#define BB 4
#define CC 3
#define FF 51
#define HO 384
#define WO 384
#define HI (HO + FF - 1)   // 434
#define WI (WO + FF - 1)   // 434

#define TILE_W 128         // output x-tile per workgroup (8 M-tiles of 16)
#define MT 16
#define TOFF 15            // tap padding: taps t in [-15, 67] addressable
#define HPAD_ROWS 83       // 15 zero + 51 real + 17 zero
#define IN_W 186           // 178 valid + pad; EVEN stride -> 8B-aligned b64 B loads
#define IN_VALID (TILE_W + FF - 1)  // 178
#define VER_W 129          // padded stride, conflict-free Ver gathers
#define NTILES 10          // ceil(153 / 16)
#define KCH 17             // K-chunks of 4 covering the 66-wide band per M-tile

typedef float v2f __attribute__((ext_vector_type(2)));
typedef float v8f __attribute__((ext_vector_type(8)));

__global__ __launch_bounds__(256) void sepconv_wmma_f32(
    const float* __restrict__ In,   // [B, C, HI, WI]
    const float* __restrict__ Ver,  // [B, F, HO, WO]
    const float* __restrict__ Hor,  // [B, F, HO, WO]
    float* __restrict__ Out)        // [B, C, HO, WO]
{
    __shared__ float sHor[HPAD_ROWS][TILE_W];        // tap table, zero outside band
    __shared__ float sVer[FF][VER_W];
    __shared__ __align__(16) float sIn[2][16][IN_W]; // double-buffered N-tile rows
    __shared__ float sOut[CC][TILE_W];

    const int tid  = threadIdx.x;
    const int lane = tid & 31;
    const int wave = tid >> 5;          // wave == M-tile index (8 waves)
    const int x0   = blockIdx.x * TILE_W;
    const int y    = blockIdx.y;
    const int n    = blockIdx.z;

    const int il    = lane & 15;        // A: row M within tile / B: column N
    const int khalf = (lane >> 4) * 2;  // lanes 0-15 -> K{0,1}, lanes 16-31 -> K{2,3}
    const int jrow  = tid >> 4;         // staging row 0..15
    const int xs    = tid & 15;

    // ---- staging helpers: thread owns row jrow, 6 float2 chunks ----
    auto load_rows = [&](int nt, v2f g[6]) {
        const int flat = nt * 16 + jrow;            // column j -> flat = fy*3 + c
        const float* src = nullptr;
        if (flat < CC * FF) {
            const int fy = flat / CC, c = flat - fy * CC;
            src = In + ((n * CC + c) * HI + (y + fy)) * WI + x0;
        }
#pragma unroll
        for (int it = 0; it < 6; ++it) {
            const int xd = xs + 16 * it;            // float2 chunk index
            v2f v = {0.0f, 0.0f};
            if (xd < IN_W / 2) {
                const int x = xd * 2;
                if (src && x < IN_VALID)            // IN_VALID (178) is even
                    v = *(const v2f*)(src + x);
            }
            g[it] = v;
        }
    };
    auto store_rows = [&](int buf, const v2f g[6]) {
#pragma unroll
        for (int it = 0; it < 6; ++it) {
            const int xd = xs + 16 * it;
            if (xd < IN_W / 2)
                *(v2f*)&sIn[buf][jrow][xd * 2] = g[it];
        }
    };

    // ---- zero output accumulator ----
    for (int i = tid; i < CC * TILE_W; i += 256)
        (&sOut[0][0])[i] = 0.0f;

    // ---- stage zero-padded Hor tap table (streamed once -> non-temporal) ----
    for (int i = tid; i < HPAD_ROWS * TILE_W; i += 256) {
        const int row = i / TILE_W, x = i - row * TILE_W;
        const int t = row - TOFF;
        float v = 0.0f;
        if (t >= 0 && t < FF)
            v = __builtin_nontemporal_load(&Hor[((n * FF + t) * HO + y) * WO + x0 + x]);
        sHor[row][x] = v;
    }
    // ---- stage Ver row (streamed once -> non-temporal) ----
    for (int i = tid; i < FF * TILE_W; i += 256) {
        const int fy = i / TILE_W, x = i - fy * TILE_W;
        sVer[fy][x] = __builtin_nontemporal_load(&Ver[((n * FF + fy) * HO + y) * WO + x0 + x]);
    }
    // ---- prologue: stage N-tile 0 ----
    {
        v2f g[6];
        load_rows(0, g);
        store_rows(0, g);
    }
    __syncthreads();

    // ---- hoist A fragments: band matrix is identical for every N-tile ----
    // A[M=il, K=khalf+c] = Hor tap t = k_off+khalf+c-il at output x = wave*16+il
    v2f afrag[KCH];
#pragma unroll
    for (int kc = 0; kc < KCH; ++kc) {
        const int k_off = kc * 4;
        afrag[kc].x = sHor[k_off + khalf + 0 - il + TOFF][wave * MT + il];
        afrag[kc].y = sHor[k_off + khalf + 1 - il + TOFF][wave * MT + il];
    }

    for (int nt = 0; nt < NTILES; ++nt) {
        const int cur = nt & 1;

        // issue next tile's global loads early; matrix pipe hides the latency
        v2f g[6];
        const bool have_next = (nt + 1 < NTILES);
        if (have_next) load_rows(nt + 1, g);

        // ---- banded WMMA: D(16x16) += A(16x4) x B(4x16), 17 K-chunks ----
        // B[K=khalf+c, N=il] = In row il at column x' = wave*16 + k_off + khalf + c
        // (il*186 + even) is even -> 8-byte aligned ds_load_b64
        v8f acc0 = {}, acc1 = {};
#pragma unroll
        for (int kc = 0; kc < KCH; kc += 2) {
            const v2f b = *(const v2f*)&sIn[cur][il][wave * MT + kc * 4 + khalf];
            acc0 = __builtin_amdgcn_wmma_f32_16x16x4_f32(
                false, afrag[kc], false, b, (short)0, acc0, false, false);
        }
#pragma unroll
        for (int kc = 1; kc < KCH; kc += 2) {
            const v2f b = *(const v2f*)&sIn[cur][il][wave * MT + kc * 4 + khalf];
            acc1 = __builtin_amdgcn_wmma_f32_16x16x4_f32(
                false, afrag[kc], false, b, (short)0, acc1, false, false);
        }
        const v8f acc = acc0 + acc1;

        // park the next tile in the other LDS buffer
        if (have_next) store_rows(cur ^ 1, g);

        // ---- vertical reduction: Out[c,x] += Ver[fy,x] * T[x,(c,fy)] ----
        // D layout: VGPR r, lanes 0-15 -> (M=r, N=lane), lanes 16-31 -> (M=r+8, N=lane-16)
        const int flat = nt * 16 + il;
        if (flat < CC * FF) {
            const int fy = flat / CC, c = flat - fy * CC;
            const int mbase = wave * MT + ((lane >> 4) << 3);
#pragma unroll
            for (int r = 0; r < 8; ++r) {
                const int xl = mbase + r;
                atomicAdd(&sOut[c][xl], acc[r] * sVer[fy][xl]); // ds_add_f32; wave owns its columns
            }
        }
        __syncthreads();   // publish next buffer + finish this tile's LDS traffic
    }

    // ---- write result (streamed once -> non-temporal) ----
    for (int i = tid; i < CC * TILE_W; i += 256) {
        const int c = i / TILE_W, x = i - c * TILE_W;
        __builtin_nontemporal_store(sOut[c][x], &Out[((n * CC + c) * HO + y) * WO + x0 + x]);
    }
}

extern "C" void kernel_launch(void* const* d_in, const int* in_sizes, int n_in,
                              void* d_out, int out_size, void* d_ws, size_t ws_size,
                              hipStream_t stream) {
    (void)in_sizes; (void)n_in; (void)out_size; (void)d_ws; (void)ws_size;
    const float* In  = (const float*)d_in[0];
    const float* Ver = (const float*)d_in[1];
    const float* Hor = (const float*)d_in[2];
    float* Out = (float*)d_out;

    dim3 grid(WO / TILE_W, HO, BB);   // (3, 384, 4)
    sepconv_wmma_f32<<<grid, 256, 0, stream>>>(In, Ver, Hor, Out);
}